// MobiusMLR_4217657884951
// MI455X (gfx1250) — compile-verified
//
#include <hip/hip_runtime.h>
#include <math.h>

typedef __attribute__((ext_vector_type(2))) float v2f;
typedef __attribute__((ext_vector_type(8))) float v8f;

#define D_DIM 128
#define K_DIM 256
#define B_DIM 4096
#define MINN 1e-15f

// ---------------------------------------------------------------------------
// Kernel 1: per-row stats. One wave32 per row (each lane loads one float4 of
// the 128-float row). Rows [0,256): p_k/a_k stats; rows [256,256+4096): |x|^2.
// ws layout (floats): [0,256) pk2 | [256,512) pa | [512,768) an | [768,4864) xn2
// ---------------------------------------------------------------------------
__global__ __launch_bounds__(256)
void mobius_stats_kernel(const float* __restrict__ x,
                         const float* __restrict__ p,
                         const float* __restrict__ a,
                         float* __restrict__ ws) {
    int wave = (int)((blockIdx.x * blockDim.x + threadIdx.x) >> 5);
    int lane = (int)(threadIdx.x & 31);
    if (wave < K_DIM) {
        const float4 pv = ((const float4*)(p + wave * D_DIM))[lane];
        const float4 av = ((const float4*)(a + wave * D_DIM))[lane];
        float sp  = pv.x * pv.x + pv.y * pv.y + pv.z * pv.z + pv.w * pv.w;
        float spa = pv.x * av.x + pv.y * av.y + pv.z * av.z + pv.w * av.w;
        float sa  = av.x * av.x + av.y * av.y + av.z * av.z + av.w * av.w;
        #pragma unroll
        for (int off = 16; off > 0; off >>= 1) {
            sp  += __shfl_xor(sp,  off, 32);
            spa += __shfl_xor(spa, off, 32);
            sa  += __shfl_xor(sa,  off, 32);
        }
        if (lane == 0) {
            ws[wave]             = sp;            // |p_k|^2
            ws[K_DIM + wave]     = spa;           // <p_k, a_k>
            ws[2 * K_DIM + wave] = sqrtf(sa);     // |a_k|
        }
    } else if (wave < K_DIM + B_DIM) {
        int r = wave - K_DIM;
        const float4 xv = ((const float4*)(x + r * D_DIM))[lane];
        float s = xv.x * xv.x + xv.y * xv.y + xv.z * xv.z + xv.w * xv.w;
        #pragma unroll
        for (int off = 16; off > 0; off >>= 1) s += __shfl_xor(s, off, 32);
        if (lane == 0) ws[3 * K_DIM + r] = s;     // |x_b|^2
    }
}

// ---------------------------------------------------------------------------
// Kernel 2: one wave32 per 16x16 output tile. Two f32 WMMA accumulators
// (PX = x.p^T, XA = x.a^T) share the A operand; D=128 swept in K=4 steps with
// V_WMMA_F32_16X16X4_F32 (exact f32 — problem is memory bound, precision is
// free). Epilogue applies the closed-form Mobius-MLR math per element.
//
// f32 16x4 A layout (ISA 7.12.2): lane m=lane&15, half=lane>>4 holds
// A[m][2*half], A[m][2*half+1] -> contiguous float2 from the row-major row.
// B (4x16) is the transpose-symmetric layout: same float2 from p/a's row.
// C/D: vgpr r -> row (r + 8*half), col (lane&15).
// ---------------------------------------------------------------------------
__global__ __launch_bounds__(256)
void mobius_mlr_wmma_kernel(const float* __restrict__ x,
                            const float* __restrict__ p,
                            const float* __restrict__ a,
                            const float* __restrict__ ws,
                            float* __restrict__ out) {
    const float* __restrict__ pk2 = ws;
    const float* __restrict__ pa  = ws + K_DIM;
    const float* __restrict__ an  = ws + 2 * K_DIM;
    const float* __restrict__ xn2 = ws + 3 * K_DIM;

    int wave = (int)((blockIdx.x * blockDim.x + threadIdx.x) >> 5);
    int lane = (int)(threadIdx.x & 31);
    const int ntk = K_DIM / 16;          // 16 tiles along K
    int b0 = (wave / ntk) * 16;
    int k0 = (wave % ntk) * 16;

    int mr   = lane & 15;
    int half = lane >> 4;

    const float* xrow = x + (size_t)(b0 + mr) * D_DIM + 2 * half;
    const float* prow = p + (size_t)(k0 + mr) * D_DIM + 2 * half;
    const float* arow = a + (size_t)(k0 + mr) * D_DIM + 2 * half;

    v8f accP = {};
    v8f accA = {};
    #pragma unroll
    for (int kk = 0; kk < D_DIM; kk += 4) {
        v2f av  = *(const v2f*)(xrow + kk);   // A tile (x), shared
        v2f bpv = *(const v2f*)(prow + kk);   // B tile (p^T)
        v2f bav = *(const v2f*)(arow + kk);   // B tile (a^T)
        accP = __builtin_amdgcn_wmma_f32_16x16x4_f32(
            false, av, false, bpv, (short)0, accP, false, false);
        accA = __builtin_amdgcn_wmma_f32_16x16x4_f32(
            false, av, false, bav, (short)0, accA, false, false);
    }

    // Per-column (class k) stats are uniform across the 8 accumulator rows.
    int k = k0 + mr;
    float x2p  = pk2[k];                 // |p_k|^2
    float pak  = pa[k];                  // <p_k, a_k>
    float an014 = an[k];
    float an32 = fmaxf(an014, MINN);     // |a_k|
    float beta = 1.0f - x2p;             // 1 - c|p|^2

    #pragma unroll
    for (int r = 0; r < 8; ++r) {
        int b = b0 + r + 8 * half;
        float PX = accP[r];
        float XA = accA[r];
        float y2 = xn2[b];                                   // |x_b|^2
        float alpha = 1.0f + y2 - 2.0f * PX;                 // 1 + 2c<-p,x> + c|x|^2
        float den = fmaxf(1.0f - 2.0f * PX + x2p * y2, MINN);
        float inv = 1.0f / den;
        float sq = (alpha * alpha * x2p - 2.0f * alpha * beta * PX
                    + beta * beta * y2) * inv * inv;         // |mpx|^2
        sq = fmaxf(sq, MINN);
        float dota = (beta * XA - alpha * pak) * inv;        // <mpx, a_k>
        float den2 = fmaxf((1.0f - sq) * an32, MINN);
        float dist = asinhf(2.0f * dota / den2);
        out[(size_t)b * K_DIM + k] = 2.0f * an32 * dist;
    }
}

extern "C" void kernel_launch(void* const* d_in, const int* in_sizes, int n_in,
                              void* d_out, int out_size, void* d_ws, size_t ws_size,
                              hipStream_t stream) {
    const float* x = (const float*)d_in[0];   // (4096, 128)
    const float* p = (const float*)d_in[1];   // (256, 128)
    const float* a = (const float*)d_in[2];   // (256, 128)
    float* out = (float*)d_out;               // (4096, 256)
    float* ws  = (float*)d_ws;                // 4864 floats used

    // Stats: (256 + 4096) rows, one wave each -> 4352 waves, 8 waves/block.
    int statsBlocks = (K_DIM + B_DIM) / 8;    // 544
    mobius_stats_kernel<<<statsBlocks, 256, 0, stream>>>(x, p, a, ws);

    // GEMM+epilogue: (4096/16)*(256/16) = 4096 tiles, 8 waves/block -> 512 blocks.
    int tiles = (B_DIM / 16) * (K_DIM / 16);
    mobius_mlr_wmma_kernel<<<tiles / 8, 256, 0, stream>>>(x, p, a, ws, out);
}